// GraphAttention_8375186227290
// MI455X (gfx1250) — compile-verified
//
#include <hip/hip_runtime.h>
#include <math.h>

typedef __attribute__((ext_vector_type(2))) float v2f;
typedef __attribute__((ext_vector_type(4))) float v4f;
typedef __attribute__((ext_vector_type(8))) float v8f;

#define NN   4096
#define INF_ 256
#define HF   512     // H * OUT_F
#define OF   128     // OUT_F
#define NHD  4
#define SLOPE 0.2f

__device__ __forceinline__ v2f ld2(const float* p) { return *(const v2f*)p; }

// ---------------- kernel 0: W (256x512) -> Wt (512x256) --------------------
__global__ void wt_kernel(const float* __restrict__ W, float* __restrict__ Wt) {
    int idx = blockIdx.x * 256 + threadIdx.x;   // 512*256 threads
    int n = idx >> 8;       // 0..511 output row
    int k = idx & 255;      // 0..255
    Wt[n * INF_ + k] = W[k * HF + n];
}

// ---------------- kernel 1: h = x @ W, stored transposed hT[col][node] -----
__global__ void __launch_bounds__(256) gemm_h_kernel(const float* __restrict__ x,
                                                     const float* __restrict__ Wt,
                                                     float* __restrict__ hT) {
    int wave = threadIdx.x >> 5;
    int lane = threadIdx.x & 31;
    int tile = blockIdx.x * 8 + wave;      // 8192 tiles: 256 M-tiles x 32 N-tiles
    int mt = tile >> 5;
    int nt = tile & 31;
    int i0 = mt * 16, n0 = nt * 16;
    int r  = lane & 15;      // M (for A) / N (for B/C) index
    int hi = lane >> 4;      // K sub-offset selector

    const float* xrow = x  + (i0 + r) * INF_ + 2 * hi;
    const float* wrow = Wt + (n0 + r) * INF_ + 2 * hi;

    v8f c = {0.f,0.f,0.f,0.f,0.f,0.f,0.f,0.f};
    for (int k = 0; k < INF_; k += 4) {
        v2f a = ld2(xrow + k);
        v2f b = ld2(wrow + k);
        c = __builtin_amdgcn_wmma_f32_16x16x4_f32(false, a, false, b,
                                                  (short)0, c, false, false);
    }
    // lane holds column n0+r, rows i0 + vr + 8*hi (vr=0..7) -> contiguous in hT
    int col = n0 + r;
    float* dst = hT + (size_t)col * NN + i0 + 8 * hi;
    v4f lo = {c[0], c[1], c[2], c[3]};
    v4f hi4 = {c[4], c[5], c[6], c[7]};
    *(v4f*)(dst)     = lo;
    *(v4f*)(dst + 4) = hi4;
}

// ---------------- kernel 2: ci/cj projections ------------------------------
__global__ void __launch_bounds__(256) cicj_kernel(const float* __restrict__ hT,
                                                   const float* __restrict__ ai,
                                                   const float* __restrict__ aj,
                                                   float* __restrict__ ciT,
                                                   float* __restrict__ cjT) {
    int n = blockIdx.x * 256 + threadIdx.x;    // node id
    for (int hd = 0; hd < NHD; hd++) {
        float si = 0.f, sj = 0.f;
        const float* hp = hT + (size_t)hd * OF * NN + n;
        for (int f = 0; f < OF; f++) {
            float hv = hp[(size_t)f * NN];          // coalesced over n
            si = fmaf(hv, ai[hd * OF + f], si);
            sj = fmaf(hv, aj[hd * OF + f], sj);
        }
        ciT[hd * NN + n] = si;
        cjT[hd * NN + n] = sj;
    }
}

// ---------------- kernel 3: fused softmax + aggregation --------------------
__global__ void __launch_bounds__(256) gat_kernel(const float* __restrict__ adj,
                                                  const float* __restrict__ hT,
                                                  const float* __restrict__ ciT,
                                                  const float* __restrict__ cjT,
                                                  const float* __restrict__ bias,
                                                  float* __restrict__ out) {
    __shared__ float red_m[256][NHD];
    __shared__ float red_l[256][NHD];
    __shared__ float m_sh[16][NHD];
    __shared__ float linv_sh[16][NHD];

    int tid = threadIdx.x;
    int i0 = blockIdx.x * 16;

    // ---- pass 1: row max m and denom l per (row, head), online softmax ----
    for (int il = 0; il < 16; il++) {
        int gi = i0 + il;
        float civ[NHD];
        for (int hd = 0; hd < NHD; hd++) civ[hd] = ciT[hd * NN + gi];

        float m[NHD], l[NHD];
        for (int hd = 0; hd < NHD; hd++) { m[hd] = -3.4e38f; l[hd] = 0.f; }

        const float* arow = adj + (size_t)gi * NN;
        for (int it = 0; it < 16; it++) {
            int j = it * 256 + tid;              // fully coalesced
            float av = arow[j];
            for (int hd = 0; hd < NHD; hd++) {
                float s = civ[hd] + cjT[hd * NN + j];
                s = (s > 0.f ? s : SLOPE * s) * av;
                float mn = fmaxf(m[hd], s);
                l[hd] = l[hd] * __expf(m[hd] - mn) + __expf(s - mn);
                m[hd] = mn;
            }
        }
        for (int hd = 0; hd < NHD; hd++) { red_m[tid][hd] = m[hd]; red_l[tid][hd] = l[hd]; }
        __syncthreads();
        for (int st = 128; st > 0; st >>= 1) {
            if (tid < st) {
                for (int hd = 0; hd < NHD; hd++) {
                    float mo = red_m[tid][hd],      lo = red_l[tid][hd];
                    float m2 = red_m[tid + st][hd], l2 = red_l[tid + st][hd];
                    float mn = fmaxf(mo, m2);
                    red_l[tid][hd] = lo * __expf(mo - mn) + l2 * __expf(m2 - mn);
                    red_m[tid][hd] = mn;
                }
            }
            __syncthreads();
        }
        if (tid < NHD) {
            m_sh[il][tid]    = red_m[0][tid];
            linv_sh[il][tid] = 1.f / red_l[0][tid];
        }
        __syncthreads();
    }

    // ---- pass 2: out[i, hd, f] = sum_j p(i,j) * h[j, hd, f] via WMMA ------
    int wave = tid >> 5, lane = tid & 31;
    int hd = wave & 3;          // head owned by this wave
    int fh = wave >> 2;         // which 64-column half
    int r  = lane & 15;
    int hi = lane >> 4;

    float m_i    = m_sh[r][hd];
    float linv_i = linv_sh[r][hd];
    float ci_i   = ciT[hd * NN + i0 + r];

    const float* adjrow = adj + (size_t)(i0 + r) * NN + 2 * hi;
    const float* cjrow  = cjT + hd * NN + 2 * hi;
    const float* hbase0 = hT + (size_t)(hd * OF + fh * 64 + 0 * 16 + r) * NN + 2 * hi;
    const float* hbase1 = hT + (size_t)(hd * OF + fh * 64 + 1 * 16 + r) * NN + 2 * hi;
    const float* hbase2 = hT + (size_t)(hd * OF + fh * 64 + 2 * 16 + r) * NN + 2 * hi;
    const float* hbase3 = hT + (size_t)(hd * OF + fh * 64 + 3 * 16 + r) * NN + 2 * hi;

    v8f acc0 = {0.f,0.f,0.f,0.f,0.f,0.f,0.f,0.f};
    v8f acc1 = acc0, acc2 = acc0, acc3 = acc0;

    for (int jb = 0; jb < NN; jb += 32) {
        // build P tile (16 x 32) for this head: each lane its 16 values
        v2f pa[8];
        #pragma unroll
        for (int kb = 0; kb < 8; kb++) {
            int j = jb + kb * 4;
            v2f av  = ld2(adjrow + j);
            v2f cjv = ld2(cjrow + j);
            #pragma unroll
            for (int t = 0; t < 2; t++) {
                float s = ci_i + cjv[t];
                s = (s > 0.f ? s : SLOPE * s) * av[t];
                pa[kb][t] = __expf(s - m_i) * av[t] * linv_i;
            }
        }
        // P (16xK) @ h (Kx64) , K = 32 in steps of 4
        #pragma unroll
        for (int kb = 0; kb < 8; kb++) {
            int j = jb + kb * 4;
            v2f b0 = ld2(hbase0 + j);
            v2f b1 = ld2(hbase1 + j);
            v2f b2 = ld2(hbase2 + j);
            v2f b3 = ld2(hbase3 + j);
            acc0 = __builtin_amdgcn_wmma_f32_16x16x4_f32(false, pa[kb], false, b0, (short)0, acc0, false, false);
            acc1 = __builtin_amdgcn_wmma_f32_16x16x4_f32(false, pa[kb], false, b1, (short)0, acc1, false, false);
            acc2 = __builtin_amdgcn_wmma_f32_16x16x4_f32(false, pa[kb], false, b2, (short)0, acc2, false, false);
            acc3 = __builtin_amdgcn_wmma_f32_16x16x4_f32(false, pa[kb], false, b3, (short)0, acc3, false, false);
        }
    }

    // epilogue: C[vr + 8*hi][r] -> out[(i0+vr+8*hi)*HF + col], add bias
    v8f accs[4] = {acc0, acc1, acc2, acc3};
    #pragma unroll
    for (int nt = 0; nt < 4; nt++) {
        int col = hd * OF + fh * 64 + nt * 16 + r;
        float bv = bias[col];
        #pragma unroll
        for (int vr = 0; vr < 8; vr++) {
            out[(size_t)(i0 + vr + 8 * hi) * HF + col] = accs[nt][vr] + bv;
        }
    }
}

// ---------------------------------------------------------------------------
extern "C" void kernel_launch(void* const* d_in, const int* in_sizes, int n_in,
                              void* d_out, int out_size, void* d_ws, size_t ws_size,
                              hipStream_t stream) {
    const float* x    = (const float*)d_in[0];
    const float* adj  = (const float*)d_in[1];
    const float* W    = (const float*)d_in[2];
    const float* a_i  = (const float*)d_in[3];
    const float* a_j  = (const float*)d_in[4];
    const float* bias = (const float*)d_in[5];
    float* out = (float*)d_out;

    float* Wt  = (float*)d_ws;                 // 512*256
    float* hT  = Wt  + HF * INF_;              // 512*4096 (8 MB)
    float* ciT = hT  + (size_t)HF * NN;        // 4*4096
    float* cjT = ciT + NHD * NN;               // 4*4096

    wt_kernel    <<<HF, 256, 0, stream>>>(W, Wt);
    gemm_h_kernel<<<1024, 256, 0, stream>>>(x, Wt, hT);
    cicj_kernel  <<<NN / 256, 256, 0, stream>>>(hT, a_i, a_j, ciT, cjT);
    gat_kernel   <<<NN / 16, 256, 0, stream>>>(adj, hT, ciT, cjT, bias, out);
}